// RecurrentPrediction_35631048688014
// MI455X (gfx1250) — compile-verified
//
#include <hip/hip_runtime.h>
#include <stdint.h>

// Problem constants (fixed by the reference)
#define RNN_B 2048
#define RNN_T 4096
#define RNN_C 3

// Tiling: 64 batch rows per block (2 waves), 64 timesteps per tile
#define BT 64
#define TT 64
#define TTP (TT + 1)          // padded LDS pitch (dwords) -> bank-conflict free
#define NT (RNN_T / TT)

#if __has_builtin(__builtin_amdgcn_tensor_load_to_lds)
#define HAVE_TDM 1
#else
#define HAVE_TDM 0
#endif

typedef __attribute__((ext_vector_type(4))) unsigned int u32x4;
typedef __attribute__((ext_vector_type(8))) int         i32x8;
typedef __attribute__((ext_vector_type(4))) int         i32x4;

__device__ __forceinline__ float fast_tanh(float x) {
#if __has_builtin(__builtin_amdgcn_tanhf)
  return __builtin_amdgcn_tanhf(x);     // gfx1250 v_tanh_f32 hardware transcendental
#else
  return tanhf(x);
#endif
}

#if HAVE_TDM
// Issue one TDM DMA: 2D tile [BT rows x TT cols] of f32 from a [*, RNN_T] tensor
// into LDS at byte offset lds_off, with LDS padding of 1 dword every TT dwords
// (pad_interval code 5 == 64 dwords, pad_amount code 0 == 1 dword) -> pitch TTP.
__device__ __forceinline__ void tdm_load_tile(const float* gsrc, unsigned lds_off) {
  unsigned long long ga = (unsigned long long)(size_t)gsrc;
  u32x4 g0;
  g0[0] = 1u;                                             // count = 1 (valid descriptor)
  g0[1] = lds_off;                                        // lds_addr [63:32]
  g0[2] = (unsigned)ga;                                   // global_addr [95:64]
  g0[3] = ((unsigned)(ga >> 32) & 0x01FFFFFFu)            // global_addr [120:96]
        | (2u << 30);                                     // type = 2 ("image")
  i32x8 g1;
  g1[0] = (int)((2u << 16)                                // data_size = 4 bytes
              | (1u << 20)                                // pad_enable
              | (5u << 22)                                // pad_interval: 64 dwords
              | (0u << 25));                              // pad_amount: 1 dword
  g1[1] = (int)(((unsigned)RNN_T & 0xFFFFu) << 16);       // tensor_dim0[15:0]
  g1[2] = (int)((((unsigned)RNN_T >> 16) & 0xFFFFu)       // tensor_dim0[31:16]
              | (((unsigned)RNN_B & 0xFFFFu) << 16));     // tensor_dim1[15:0]
  g1[3] = (int)((((unsigned)RNN_B >> 16) & 0xFFFFu)       // tensor_dim1[31:16]
              | ((unsigned)TT << 16));                    // tile_dim0 = TT
  g1[4] = (int)BT;                                        // tile_dim1 = BT rows (tile_dim2=0)
  g1[5] = (int)RNN_T;                                     // tensor_dim0_stride[31:0]
  g1[6] = 0;                                              // stride hi / dim1_stride lo
  g1[7] = 0;
  i32x4 z4; z4[0] = 0; z4[1] = 0; z4[2] = 0; z4[3] = 0;
  i32x8 z8; z8[0] = 0; z8[1] = 0; z8[2] = 0; z8[3] = 0;
  z8[4] = 0; z8[5] = 0; z8[6] = 0; z8[7] = 0;
  // clang-23 / therock-10.0 arity: (g0, g1, g2, g3, g4, cpol)
  __builtin_amdgcn_tensor_load_to_lds(g0, g1, z4, z4, z8, 0);
}
__device__ __forceinline__ void wait_tensor0() {
  __builtin_amdgcn_s_wait_tensorcnt(0);
}
__device__ __forceinline__ void wait_tensor1() {
  __builtin_amdgcn_s_wait_tensorcnt(1);
}
#endif

__global__ __launch_bounds__(BT) void rnn_scan_kernel(
    const float* __restrict__ h,     // [B, T]
    const float* __restrict__ Wih,   // [C, 2]
    const float* __restrict__ Whh,   // [C, C]
    const float* __restrict__ bih,   // [C]
    const float* __restrict__ bhh,   // [C]
    float* __restrict__ out)         // [B, C, T]
{
  __shared__ float lds_in[2][BT * TTP];           // double-buffered input tiles
  __shared__ float lds_out[RNN_C * BT * TTP];     // output staging tile [c][r][t]

  const int tid = threadIdx.x;                    // 0..63 == batch row within block
  const int b0  = blockIdx.x * BT;

  // ---- fold weights (uniform scalar loads) ----
  // pre_c = e_c + u_c*h_t + v_c*h_prev  where u=Wih[c][0]+Wih[c][1], v=-Wih[c][1]
  const float u0 = Wih[0] + Wih[1], v0 = -Wih[1];
  const float u1 = Wih[2] + Wih[3], v1 = -Wih[3];
  const float u2 = Wih[4] + Wih[5], v2 = -Wih[5];
  const float e0 = bih[0] + bhh[0];
  const float e1 = bih[1] + bhh[1];
  const float e2 = bih[2] + bhh[2];
  const float w00 = Whh[0], w01 = Whh[1], w02 = Whh[2];
  const float w10 = Whh[3], w11 = Whh[4], w12 = Whh[5];
  const float w20 = Whh[6], w21 = Whh[7], w22 = Whh[8];

  float y0 = 0.f, y1 = 0.f, y2 = 0.f, hprev = 0.f;   // recurrence carry

  float* orow0 = &lds_out[(0 * BT + tid) * TTP];
  float* orow1 = &lds_out[(1 * BT + tid) * TTP];
  float* orow2 = &lds_out[(2 * BT + tid) * TTP];

#if HAVE_TDM
  const bool issuer = (tid < 32);                 // wave 0 drives the TDM
  const unsigned lds_in_off0 = (unsigned)(size_t)(void*)&lds_in[0][0];
  const unsigned lds_in_off1 = (unsigned)(size_t)(void*)&lds_in[1][0];
  if (issuer) {
    tdm_load_tile(h + (size_t)b0 * RNN_T, lds_in_off0);  // prefetch tile 0
  }
#endif

  for (int k = 0; k < NT; ++k) {
#if HAVE_TDM
    if (issuer) {
      if (k + 1 < NT) {
        // prefetch next tile into the other buffer, then wait for current tile
        tdm_load_tile(h + (size_t)b0 * RNN_T + (size_t)(k + 1) * TT,
                      ((k + 1) & 1) ? lds_in_off1 : lds_in_off0);
        wait_tensor1();              // TDM loads complete in order: tile k is done
      } else {
        wait_tensor0();
      }
    }
#else
    {
      // Fallback: cooperative coalesced load of tile k (lanes vary t)
      float* buf = lds_in[k & 1];
      for (int idx = tid; idx < BT * TT; idx += BT) {
        int t = idx & (TT - 1);
        int r = idx >> 6;            // TT == 64
        buf[r * TTP + t] = h[(size_t)(b0 + r) * RNN_T + (size_t)k * TT + t];
      }
    }
#endif
    __syncthreads();                 // tile k visible to all waves

    // ---- sequential recurrence over this tile (registers only) ----
    const float* irow = &lds_in[k & 1][tid * TTP];
#pragma unroll 8
    for (int t = 0; t < TT; ++t) {
      const float ht = irow[t];
      float p0 = __builtin_fmaf(u0, ht, __builtin_fmaf(v0, hprev, e0));
      float p1 = __builtin_fmaf(u1, ht, __builtin_fmaf(v1, hprev, e1));
      float p2 = __builtin_fmaf(u2, ht, __builtin_fmaf(v2, hprev, e2));
      float s0 = __builtin_fmaf(w00, y0, __builtin_fmaf(w01, y1, __builtin_fmaf(w02, y2, p0)));
      float s1 = __builtin_fmaf(w10, y0, __builtin_fmaf(w11, y1, __builtin_fmaf(w12, y2, p1)));
      float s2 = __builtin_fmaf(w20, y0, __builtin_fmaf(w21, y1, __builtin_fmaf(w22, y2, p2)));
      y0 = fast_tanh(s0);
      y1 = fast_tanh(s1);
      y2 = fast_tanh(s2);
      hprev = ht;
      orow0[t] = y0;
      orow1[t] = y1;
      orow2[t] = y2;
    }
    __syncthreads();                 // out tile complete

    // ---- cooperative coalesced store: lanes vary t ----
    for (int idx = tid; idx < RNN_C * BT * TT; idx += BT) {
      int t  = idx & (TT - 1);
      int rc = idx >> 6;             // TT == 64
      int r  = rc & (BT - 1);
      int c  = rc >> 6;              // BT == 64
      out[(size_t)(b0 + r) * (RNN_C * RNN_T) + (size_t)c * RNN_T + (size_t)k * TT + t] =
          lds_out[(c * BT + r) * TTP + t];
    }
    __syncthreads();                 // lds_out + input buffer reusable
  }
}

extern "C" void kernel_launch(void* const* d_in, const int* in_sizes, int n_in,
                              void* d_out, int out_size, void* d_ws, size_t ws_size,
                              hipStream_t stream) {
  (void)in_sizes; (void)n_in; (void)out_size; (void)d_ws; (void)ws_size;
  const float* h   = (const float*)d_in[0];
  const float* Wih = (const float*)d_in[1];
  const float* Whh = (const float*)d_in[2];
  const float* bih = (const float*)d_in[3];
  const float* bhh = (const float*)d_in[4];
  float* out = (float*)d_out;

  dim3 grid(RNN_B / BT);   // 32 blocks
  dim3 block(BT);          // 64 threads (2 wave32)
  rnn_scan_kernel<<<grid, block, 0, stream>>>(h, Wih, Whh, bih, bhh, out);
}